// deepGNN_32873679684165
// MI455X (gfx1250) — compile-verified
//
#include <hip/hip_runtime.h>

// ---------------------------------------------------------------------------
// deepGNN on MI455X (gfx1250), f32 end-to-end using V_WMMA_F32_16X16X4_F32.
//
// Roofline: ~145 G MACs f32 vs. memory. HBM traffic already minimized
// (~0.45 GB: edge_attr read once, e lives only in LDS, aggr L2-resident).
// This revision attacks the next bottleneck: weight re-streaming from L2.
// Each block now processes 64 rows (4 m-tiles); a wave holds each weight
// (B) fragment in registers and issues 4 independent WMMAs with it, cutting
// L2 weight traffic 4x and giving 4-deep ILP behind every global_load_b64.
// ---------------------------------------------------------------------------

typedef float v2f __attribute__((ext_vector_type(2)));
typedef float v8f __attribute__((ext_vector_type(8)));

#define SLOPE  0.01f
#define LN_EPS 1e-5f
#define DIM    128
#define NNODES 50000
#define NEDGES 640000

// ---------------------------------------------------------------------------
// One GEMM stage of an MLP over MT 16-row tiles, whole 256-thread block:
//   out_lds[16*MT][outPitch] = act( in_lds[16*MT][inPitch] @ W^T + bias )
// W is torch-Linear layout [Nvalid][Kin] row-major in global memory.
// Kpad: 4-aligned >= Kin (in_lds must be zero beyond Kin).
// Npad: 16-aligned >= Nvalid (padded outputs are written as exact zeros).
//
// WMMA f32 16x16x4 fragment layout (CDNA5 ISA 7.12.2):
//   A (16x4):  lane L -> row (L&15), K = k0 + (L>>4)*2 + {0,1}   (2 VGPRs)
//   B (4x16):  lane L -> col (L&15), K = k0 + (L>>4)*2 + {0,1}   (mirrored)
//   C/D (16x16): vgpr v, lane L -> row v + 8*(L>>4), col (L&15)
// Each wave owns one n-tile; B fragments are loaded once per k-step and
// reused across the MT m-tiles (MT independent accumulators).
// ---------------------------------------------------------------------------
template <int MT>
__device__ __forceinline__ void mlp_stage(
    const float* __restrict__ W, const float* __restrict__ bias,
    const float* __restrict__ in_lds, int inPitch, int Kin, int Kpad,
    float* __restrict__ out_lds, int outPitch, int Nvalid, int Npad,
    bool leaky)
{
    const int tid    = threadIdx.x;
    const int lane   = tid & 31;
    const int wave   = tid >> 5;
    const int nwaves = blockDim.x >> 5;
    const int r       = lane & 15;          // A-row / B-col within tile
    const int kg      = (lane >> 4) << 1;   // K sub-offset for this half-wave
    const int rowBase = (lane >> 4) << 3;   // D rows 0..7 or 8..15

    for (int nt = wave; nt * 16 < Npad; nt += nwaves) {
        const int n = nt * 16 + r;
        const float* wrow = W + (size_t)n * (size_t)Kin;  // deref only if n<Nvalid
        v8f acc[MT];
#pragma unroll
        for (int mt = 0; mt < MT; ++mt)
            acc[mt] = (v8f){0.f, 0.f, 0.f, 0.f, 0.f, 0.f, 0.f, 0.f};

        for (int k = 0; k < Kpad; k += 4) {
            v2f b;
            b.x = 0.f; b.y = 0.f;
            if (n < Nvalid && (k + kg) < Kin) {  // all K dims here are even
                b.x = wrow[k + kg];
                b.y = wrow[k + kg + 1];
            }
#pragma unroll
            for (int mt = 0; mt < MT; ++mt) {
                v2f a;
                a.x = in_lds[(mt * 16 + r) * inPitch + k + kg];
                a.y = in_lds[(mt * 16 + r) * inPitch + k + kg + 1];
                acc[mt] = __builtin_amdgcn_wmma_f32_16x16x4_f32(
                    false, a, false, b, (short)0, acc[mt], false, false);
            }
        }
        const float bn = (n < Nvalid) ? bias[n] : 0.f;
#pragma unroll
        for (int mt = 0; mt < MT; ++mt) {
#pragma unroll
            for (int v = 0; v < 8; ++v) {
                float val = acc[mt][v] + bn;
                if (leaky) val = (val > 0.f) ? val : SLOPE * val;
                out_lds[(mt * 16 + rowBase + v) * outPitch + n] = val;
            }
        }
    }
}

// LayerNorm over Dv features for 16 rows in LDS; 256 threads (16 per row).
__device__ __forceinline__ void layernorm16(
    float* __restrict__ h, int pitch, int Dv,
    const float* __restrict__ g, const float* __restrict__ bt,
    float* __restrict__ red /* 16*17 floats */)
{
    const int tid = threadIdx.x;
    const int row = tid >> 4;
    const int j   = tid & 15;
    float* rrow = red + row * 17;

    float s = 0.f;
    for (int c = j; c < Dv; c += 16) s += h[row * pitch + c];
    rrow[j] = s;
    __syncthreads();
    float mu = 0.f;
#pragma unroll
    for (int q = 0; q < 16; ++q) mu += rrow[q];
    mu *= (1.0f / Dv);
    __syncthreads();

    float s2 = 0.f;
    for (int c = j; c < Dv; c += 16) {
        float d = h[row * pitch + c] - mu;
        s2 += d * d;
    }
    rrow[j] = s2;
    __syncthreads();
    float var = 0.f;
#pragma unroll
    for (int q = 0; q < 16; ++q) var += rrow[q];
    const float inv = rsqrtf(var * (1.0f / Dv) + LN_EPS);
    __syncthreads();

    for (int c = j; c < Dv; c += 16) {
        float v = (h[row * pitch + c] - mu) * inv;
        h[row * pitch + c] = v * g[c] + bt[c];
    }
    __syncthreads();
}

template <int MT>
__device__ __forceinline__ void layernorm_mt(
    float* h, int pitch, int Dv,
    const float* g, const float* bt, float* red)
{
#pragma unroll 1
    for (int mt = 0; mt < MT; ++mt)
        layernorm16(h + mt * 16 * pitch, pitch, Dv, g, bt, red);
}

// Load 16*MT rows x 128 cols from global into LDS (float4; zero-fill rows
// past nRows so downstream math stays finite; pitch multiple of 4).
template <int MT>
__device__ __forceinline__ void load_tile128(
    const float* __restrict__ src, long r0, long nRows,
    float* __restrict__ dst, int pitch)
{
    const float4* __restrict__ s4 = (const float4*)src;
    for (int idx = threadIdx.x; idx < MT * 16 * 32; idx += blockDim.x) {
        const int row = idx >> 5, c4 = idx & 31;
        float4 v = {0.f, 0.f, 0.f, 0.f};
        if (r0 + row < nRows) v = s4[(r0 + row) * 32 + c4];
        *(float4*)&dst[row * pitch + c4 * 4] = v;
    }
}

template <int MT>
__device__ __forceinline__ void store_tile128(
    float* __restrict__ dst, long r0, long nRows,
    const float* __restrict__ src, int pitch)
{
    for (int idx = threadIdx.x; idx < MT * 16 * 32; idx += blockDim.x) {
        const int row = idx >> 5, c4 = idx & 31;
        if (r0 + row < nRows)
            *(float4*)&dst[((r0 + row) << 7) + c4 * 4] =
                *(const float4*)&src[row * pitch + c4 * 4];
    }
}

// ------------------------- kernels -----------------------------------------

__global__ void zero_kernel(float* __restrict__ p, size_t n)
{
    size_t i = (size_t)blockIdx.x * blockDim.x + threadIdx.x;
    const size_t stride = (size_t)gridDim.x * blockDim.x;
    for (; i < n; i += stride) p[i] = 0.f;
}

// Y = LayerNorm(MLP3(X)), all widths 128: node encoder. 64 rows/block.
__global__ __launch_bounds__(256) void fused_mlp3_ln_kernel(
    const float* __restrict__ X,
    const float* __restrict__ W0, const float* __restrict__ b0,
    const float* __restrict__ W1, const float* __restrict__ b1,
    const float* __restrict__ W2, const float* __restrict__ b2,
    const float* __restrict__ g,  const float* __restrict__ bt,
    float* __restrict__ Y)
{
    __shared__ float bufA[64 * 132];
    __shared__ float bufB[64 * 132];
    __shared__ float red[16 * 17];
    const long r0 = (long)blockIdx.x * 64;

    load_tile128<4>(X, r0, NNODES, bufA, 132);
    __syncthreads();
    mlp_stage<4>(W0, b0, bufA, 132, 128, 128, bufB, 132, 128, 128, true);
    __syncthreads();
    mlp_stage<4>(W1, b1, bufB, 132, 128, 128, bufA, 132, 128, 128, true);
    __syncthreads();
    mlp_stage<4>(W2, b2, bufA, 132, 128, 128, bufB, 132, 128, 128, false);
    __syncthreads();
    layernorm_mt<4>(bufB, 132, 128, g, bt, red);

    store_tile128<4>(Y, r0, NNODES, bufB, 132);
}

// -------------------- fused edge mega-kernel --------------------------------
// Per 64-edge tile: e = LN(MLP3(edge_attr)) lives only in LDS; then for each
// of the 3 GNN layers msg_l = MLP3_l(e) and scatter-add into aggr_l.
struct EdgeP {
    // [0..5] enc_edge W0,b0,W1,b1,W2,b2 | [6..7] enc_edge_ln g,b
    // [8+6l .. 13+6l] msg_l W0,b0,W1,b1,W2,b2   (l = 0..2)
    const float* q[26];
};

__global__ __launch_bounds__(256) void edge_mega_kernel(
    const float* __restrict__ EA, const int* __restrict__ dst, EdgeP P,
    float* __restrict__ aggr0, float* __restrict__ aggr1,
    float* __restrict__ aggr2)
{
    __shared__ float ebuf[64 * 132];   // persistent encoded-edge tile
    __shared__ float bufA[64 * 132];
    __shared__ float bufB[64 * 132];
    __shared__ float red[16 * 17];
    const long r0 = (long)blockIdx.x * 64;

    // edge encoder: edge_attr -> e (LDS only)
    load_tile128<4>(EA, r0, NEDGES, bufA, 132);
    __syncthreads();
    mlp_stage<4>(P.q[0], P.q[1], bufA, 132, 128, 128, bufB, 132, 128, 128, true);
    __syncthreads();
    mlp_stage<4>(P.q[2], P.q[3], bufB, 132, 128, 128, bufA, 132, 128, 128, true);
    __syncthreads();
    mlp_stage<4>(P.q[4], P.q[5], bufA, 132, 128, 128, ebuf, 132, 128, 128, false);
    __syncthreads();
    layernorm_mt<4>(ebuf, 132, 128, P.q[6], P.q[7], red);

    float* aggrs[3] = {aggr0, aggr1, aggr2};
#pragma unroll 1
    for (int l = 0; l < 3; ++l) {
        const float* const* m = &P.q[8 + 6 * l];
        mlp_stage<4>(m[0], m[1], ebuf, 132, 128, 128, bufA, 132, 128, 128, true);
        __syncthreads();
        mlp_stage<4>(m[2], m[3], bufA, 132, 128, 128, bufB, 132, 128, 128, true);
        __syncthreads();
        mlp_stage<4>(m[4], m[5], bufB, 132, 128, 128, bufA, 132, 128, 128, false);
        __syncthreads();
        float* __restrict__ ag = aggrs[l];
        for (int idx = threadIdx.x; idx < 64 * 128; idx += blockDim.x) {
            const int row = idx >> 7, col = idx & 127;
            const long node = (long)dst[r0 + row];
            atomicAdd(&ag[(node << 7) + col], bufA[row * 132 + col]);
        }
        __syncthreads();  // bufA reused by next layer's stage 1
    }
}

// h = LN(concat[y, aggr]); out = MLP3(h) 256->214->172->128; y += LN(out)
// 64 rows/block; h2 aliases the dead cbuf region to fit 2 blocks per WGP.
__global__ __launch_bounds__(256) void node_update_kernel(
    float* __restrict__ Y, const float* __restrict__ aggr,
    const float* __restrict__ gc, const float* __restrict__ bc,
    const float* __restrict__ W0, const float* __restrict__ b0,
    const float* __restrict__ W1, const float* __restrict__ b1,
    const float* __restrict__ W2, const float* __restrict__ b2,
    const float* __restrict__ go, const float* __restrict__ bo)
{
    // cbuf 64*260 | h1 64*228 | h3 64*132 | red 16*17   (h2 = cbuf region)
    __shared__ float smem[64 * 260 + 64 * 228 + 64 * 132 + 16 * 17];
    float* cbuf = smem;                       // K = 256 concat (pitch 260)
    float* h1   = smem + 64 * 260;            // 214 valid, padded to 224
    float* h2   = smem;                       // 172 valid, padded to 176
    float* h3   = smem + 64 * 260 + 64 * 228; // 128 (pitch 132)
    float* red  = h3 + 64 * 132;
    const long r0 = (long)blockIdx.x * 64;

    for (int idx = threadIdx.x; idx < 64 * 32; idx += blockDim.x) {
        const int row = idx >> 5, c4 = idx & 31;
        float4 yv = {0.f, 0.f, 0.f, 0.f}, av = {0.f, 0.f, 0.f, 0.f};
        if (r0 + row < NNODES) {
            yv = *(const float4*)&Y[((r0 + row) << 7) + c4 * 4];
            av = *(const float4*)&aggr[((r0 + row) << 7) + c4 * 4];
        }
        *(float4*)&cbuf[row * 260 + c4 * 4]       = yv;
        *(float4*)&cbuf[row * 260 + 128 + c4 * 4] = av;
    }
    __syncthreads();
    layernorm_mt<4>(cbuf, 260, 256, gc, bc, red);

    mlp_stage<4>(W0, b0, cbuf, 260, 256, 256, h1, 228, 214, 224, true);
    __syncthreads();
    mlp_stage<4>(W1, b1, h1, 228, 214, 216, h2, 180, 172, 176, true);  // h2 over cbuf
    __syncthreads();
    mlp_stage<4>(W2, b2, h2, 180, 172, 172, h3, 132, 128, 128, false);
    __syncthreads();
    layernorm_mt<4>(h3, 132, 128, go, bo, red);

    for (int idx = threadIdx.x; idx < 64 * 32; idx += blockDim.x) {
        const int row = idx >> 5, c4 = idx & 31;
        if (r0 + row < NNODES) {
            float4 yv = *(const float4*)&Y[((r0 + row) << 7) + c4 * 4];
            const float4 hv = *(const float4*)&h3[row * 132 + c4 * 4];
            yv.x += hv.x; yv.y += hv.y; yv.z += hv.z; yv.w += hv.w;
            *(float4*)&Y[((r0 + row) << 7) + c4 * 4] = yv;  // residual
        }
    }
}

// out = MLP3(y): 128 -> 86 -> 44 -> 2.  64 rows/block.
__global__ __launch_bounds__(256) void decode_kernel(
    const float* __restrict__ Y,
    const float* __restrict__ W0, const float* __restrict__ b0,
    const float* __restrict__ W1, const float* __restrict__ b1,
    const float* __restrict__ W2, const float* __restrict__ b2,
    float* __restrict__ out)
{
    __shared__ float bufA[64 * 132];
    __shared__ float h1[64 * 100];  // 86 valid, padded to 96
    __shared__ float h2[64 * 52];   // 44 valid, padded to 48
    __shared__ float h3[64 * 20];   // 2 valid, padded to 16
    const long r0 = (long)blockIdx.x * 64;

    load_tile128<4>(Y, r0, NNODES, bufA, 132);
    __syncthreads();
    mlp_stage<4>(W0, b0, bufA, 132, 128, 128, h1, 100, 86, 96, true);
    __syncthreads();
    mlp_stage<4>(W1, b1, h1, 100, 86, 88, h2, 52, 44, 48, true);
    __syncthreads();
    mlp_stage<4>(W2, b2, h2, 52, 44, 44, h3, 20, 2, 16, false);
    __syncthreads();

    for (int idx = threadIdx.x; idx < 64 * 2; idx += blockDim.x) {
        const int row = idx >> 1, col = idx & 1;
        if (r0 + row < NNODES)
            out[(r0 + row) * 2 + col] = h3[row * 20 + col];
    }
}

// ------------------------- host launcher ------------------------------------
// d_in order (setup_inputs insertion order, depth-first leaves):
//  0 x | 1 edge_index | 2 edge_attr
//  3..8  enc W0,b0,W1,b1,W2,b2 | 9,10 enc_ln g,b
// 11..16 enc_edge            | 17,18 enc_edge_ln g,b
// per layer l (base = 19+16*l):
//  +0..5 msg W/b | +6,7 ln_cat g,b | +8..13 upd W/b | +14,15 ln_out g,b
// 67..72 dec W0,b0,W1,b1,W2,b2

extern "C" void kernel_launch(void* const* d_in, const int* in_sizes, int n_in,
                              void* d_out, int out_size, void* d_ws, size_t ws_size,
                              hipStream_t stream)
{
    (void)in_sizes; (void)n_in; (void)out_size; (void)ws_size;

    const float* x     = (const float*)d_in[0];
    const int*   eidx  = (const int*)d_in[1];
    const float* eattr = (const float*)d_in[2];
    const int*   dst   = eidx + NEDGES;  // edge_index[1]

    auto p = [&](int i) { return (const float*)d_in[i]; };

    // workspace: y[N,128] | aggr0 | aggr1 | aggr2   (~102 MB)
    float* y     = (float*)d_ws;
    float* aggr0 = y + (size_t)NNODES * DIM;
    float* aggr1 = aggr0 + (size_t)NNODES * DIM;
    float* aggr2 = aggr1 + (size_t)NNODES * DIM;

    const int nodeGrid = (NNODES + 63) / 64;   // 782 (last block row-guarded)
    const int edgeGrid = NEDGES / 64;          // 10000 exact

    // aggr buffers are contiguous: zero all three in one pass
    zero_kernel<<<2048, 256, 0, stream>>>(aggr0, (size_t)3 * NNODES * DIM);

    fused_mlp3_ln_kernel<<<nodeGrid, 256, 0, stream>>>(
        x, p(3), p(4), p(5), p(6), p(7), p(8), p(9), p(10), y);

    EdgeP P;
    for (int i = 0; i < 8; ++i) P.q[i] = p(11 + i);  // enc_edge + its LN
    for (int l = 0; l < 3; ++l)
        for (int i = 0; i < 6; ++i) P.q[8 + 6 * l + i] = p(19 + 16 * l + i);
    edge_mega_kernel<<<edgeGrid, 256, 0, stream>>>(
        eattr, dst, P, aggr0, aggr1, aggr2);

    float* aggrs[3] = {aggr0, aggr1, aggr2};
    for (int l = 0; l < 3; ++l) {
        const int b = 19 + 16 * l;
        node_update_kernel<<<nodeGrid, 256, 0, stream>>>(
            y, aggrs[l], p(b + 6), p(b + 7),
            p(b + 8), p(b + 9), p(b + 10), p(b + 11), p(b + 12), p(b + 13),
            p(b + 14), p(b + 15));
    }

    decode_kernel<<<nodeGrid, 256, 0, stream>>>(
        y, p(67), p(68), p(69), p(70), p(71), p(72), (float*)d_out);
}